// CrossAttention_42365557408181
// MI455X (gfx1250) — compile-verified
//
#include <hip/hip_runtime.h>
#include <hip/hip_bf16.h>

typedef __attribute__((ext_vector_type(2))) float v2f;
typedef __attribute__((ext_vector_type(4))) float v4f;
typedef __attribute__((ext_vector_type(8))) float v8f;
typedef __attribute__((ext_vector_type(4))) int   v4i;

constexpr int   kB  = 2;
constexpr int   kN  = 4096;
constexpr int   kM  = 1024;
constexpr int   kQD = 512;
constexpr int   kCD = 768;
constexpr int   kH  = 8;
constexpr int   kDH = 64;
constexpr int   kHD = kH * kDH;      // 512
constexpr float kScale = 0.125f;     // DH^-0.5

#if __has_builtin(__builtin_amdgcn_global_load_async_to_lds_b128) && \
    __has_builtin(__builtin_amdgcn_s_wait_asynccnt)
#define ASYNC_COPY 1
#else
#define ASYNC_COPY 0
#endif

// 16-byte global -> LDS copy (async on CDNA5 if the builtin is available).
__device__ __forceinline__ void copy16B(const float* g, float* l) {
#if ASYNC_COPY
  __builtin_amdgcn_global_load_async_to_lds_b128((v4i*)g, (v4i*)l, 0, 0);
#else
  *(v4f*)l = *(const v4f*)g;
#endif
}

template <int ALLOW>
__device__ __forceinline__ void copy_wait() {
#if ASYNC_COPY
  __builtin_amdgcn_s_wait_asynccnt(ALLOW);
#endif
}

// ---------------------------------------------------------------------------
// fp32 WMMA GEMM: C[Mr,Nn] = A[Mr,K] @ W[K,Nn] (+ bias[Nn] if given).
// Block (8 waves) computes a 128x64 tile; K staged in 16-wide chunks through
// double-buffered LDS with async-to-LDS copies (3 b128 ops per thread/chunk).
// A-frag (16x4 f32): lane<16 -> row=lane, {K0,K1}; lane>=16 -> row=lane-16, {K2,K3}.
// B-frag (4x16 f32): lane<16 -> col=lane, {K0,K1}; lane>=16 -> col=lane-16, {K2,K3}.
// C/D (16x16 f32):   vgpr r -> row = r + (lane>=16 ? 8 : 0), col = lane&15.
// ---------------------------------------------------------------------------
__global__ __launch_bounds__(256)
void gemm_wmma_f32(const float* __restrict__ A, const float* __restrict__ W,
                   const float* __restrict__ bias, float* __restrict__ C,
                   int Mr, int K, int Nn) {
  __shared__ float As[2][128][20];   // 20 floats/row: 16B-aligned, conflict-free
  __shared__ float Ws[2][16][64];

  const int tid  = threadIdx.x;
  const int wv   = tid >> 5;
  const int lane = tid & 31;
  const int lrow = lane & 15;
  const int lhi  = lane >> 4;

  const int colGroups = Nn / 64;
  const int rb = blockIdx.x / colGroups;
  const int cg = blockIdx.x % colGroups;
  const int r0 = rb * 128;
  const int c0 = cg * 64;

  // Stage one K-chunk (A: 128x16, W: 16x64) into LDS buffer `buf`.
  auto stage = [&](int buf, int kc) {
#pragma unroll
    for (int it = 0; it < 2; ++it) {
      const int idx = tid + it * 256;        // 512 quarter-rows of A
      const int row = idx >> 2;
      const int q   = idx & 3;
      copy16B(A + (size_t)(r0 + row) * K + kc + q * 4, &As[buf][row][q * 4]);
    }
    copy16B(W + (size_t)(kc + (tid >> 4)) * Nn + c0 + (tid & 15) * 4,
            &Ws[buf][tid >> 4][(tid & 15) * 4]);
  };

  v8f acc[4] = {};
  const int nCh = K / 16;

  stage(0, 0);
  for (int ch = 0; ch < nCh; ++ch) {
    const int buf = ch & 1;
    if (ch + 1 < nCh) {
      stage(buf ^ 1, (ch + 1) * 16);   // prefetch next chunk (async)
      copy_wait<3>();                  // allow next chunk's 3 ops in flight
    } else {
      copy_wait<0>();
    }
    __syncthreads();                   // chunk `ch` resident for all waves

#pragma unroll
    for (int d = 0; d < 4; ++d) {
      v2f a;
      a.x = As[buf][wv * 16 + lrow][d * 4 + lhi * 2 + 0];
      a.y = As[buf][wv * 16 + lrow][d * 4 + lhi * 2 + 1];
#pragma unroll
      for (int t = 0; t < 4; ++t) {
        v2f b;
        b.x = Ws[buf][d * 4 + lhi * 2 + 0][t * 16 + lrow];
        b.y = Ws[buf][d * 4 + lhi * 2 + 1][t * 16 + lrow];
        acc[t] = __builtin_amdgcn_wmma_f32_16x16x4_f32(
            false, a, false, b, (short)0, acc[t], false, false);
      }
    }
    __syncthreads();                   // done reading `buf` before re-stage
  }

  const int wr0 = r0 + wv * 16;
#pragma unroll
  for (int t = 0; t < 4; ++t) {
    const int col = c0 + t * 16 + lrow;
    const float bv = bias ? bias[col] : 0.0f;
#pragma unroll
    for (int r = 0; r < 8; ++r)
      C[(size_t)(wr0 + r + lhi * 8) * Nn + col] = acc[t][r] + bv;
  }
}

// ---------------------------------------------------------------------------
// Fused flash-style cross-attention.
// Grid: (N/16, B); block: 256 threads = 8 waves, wave h owns head h.
// Two query passes (projected x, then adapt) share K/V; outputs summed.
// Pass 0 also emits the head-mean of the scaled logits ([B,N,M]).
// ---------------------------------------------------------------------------
__global__ __launch_bounds__(256)
void attn_fused(const float* __restrict__ qg, const float* __restrict__ adaptg,
                const float* __restrict__ kg, const float* __restrict__ vg,
                float* __restrict__ og, float* __restrict__ probs) {
  __shared__ float s_sim[kH][16][17];   // scaled logits tile per head
  __shared__ float s_p[kH][16][17];     // exp(logit - rowmax) tile per head
  __shared__ float s_f[kH][16];         // per-row broadcast factors

  const int h    = threadIdx.x >> 5;
  const int lane = threadIdx.x & 31;
  const int lrow = lane & 15;
  const int lhi  = lane >> 4;
  const int n0   = blockIdx.x * 16;
  const int bI   = blockIdx.y;

  v8f res[4] = {};

  for (int pass = 0; pass < 2; ++pass) {
    const float* qsrc = pass ? adaptg : qg;

    v2f qa[16];
    const float* qrow = qsrc + ((size_t)(bI * kN + n0 + lrow) * kHD + h * kDH);
#pragma unroll
    for (int d = 0; d < 16; ++d) {
      qa[d].x = qrow[d * 4 + lhi * 2 + 0];
      qa[d].y = qrow[d * 4 + lhi * 2 + 1];
    }

    float mrow  = -3.0e38f;
    float denom = 0.0f;
    v8f   ot[4] = {};

    for (int mt = 0; mt < kM / 16; ++mt) {
      const int m0 = mt * 16;

      v8f c = {};
      const float* krow = kg + ((size_t)(bI * kM + m0 + lrow) * kHD + h * kDH);
#pragma unroll
      for (int d = 0; d < 16; ++d) {
        v2f bf;
        bf.x = krow[d * 4 + lhi * 2 + 0];
        bf.y = krow[d * 4 + lhi * 2 + 1];
        c = __builtin_amdgcn_wmma_f32_16x16x4_f32(
            false, qa[d], false, bf, (short)0, c, false, false);
      }
#pragma unroll
      for (int r = 0; r < 8; ++r)
        s_sim[h][r + lhi * 8][lrow] = c[r] * kScale;

      if (pass == 0) {
        __syncthreads();
        const int row = threadIdx.x >> 4;
        const int col = threadIdx.x & 15;
        float s = 0.0f;
#pragma unroll
        for (int hh = 0; hh < kH; ++hh) s += s_sim[hh][row][col];
        probs[(size_t)(bI * kN + n0 + row) * kM + m0 + col] = s * (1.0f / kH);
        __syncthreads();
      }

      float sv[8];
#pragma unroll
      for (int j = 0; j < 8; ++j) sv[j] = s_sim[h][lrow][lhi * 8 + j];
      float tmax = sv[0];
#pragma unroll
      for (int j = 1; j < 8; ++j) tmax = fmaxf(tmax, sv[j]);
      tmax = fmaxf(tmax, __shfl_xor(tmax, 16, 32));
      const float newm = fmaxf(mrow, tmax);
      const float corr = __expf(mrow - newm);
      float psum = 0.0f;
#pragma unroll
      for (int j = 0; j < 8; ++j) {
        const float p = __expf(sv[j] - newm);
        s_p[h][lrow][lhi * 8 + j] = p;
        psum += p;
      }
      psum += __shfl_xor(psum, 16, 32);
      denom = denom * corr + psum;
      mrow  = newm;
      s_f[h][lrow] = corr;

      float cf[8];
#pragma unroll
      for (int r = 0; r < 8; ++r) cf[r] = s_f[h][r + lhi * 8];
#pragma unroll
      for (int t = 0; t < 4; ++t)
#pragma unroll
        for (int r = 0; r < 8; ++r) ot[t][r] *= cf[r];

#pragma unroll
      for (int d2 = 0; d2 < 4; ++d2) {
        v2f af;
        af.x = s_p[h][lrow][d2 * 4 + lhi * 2 + 0];
        af.y = s_p[h][lrow][d2 * 4 + lhi * 2 + 1];
        const float* vrow =
            vg + ((size_t)(bI * kM + m0 + d2 * 4 + lhi * 2) * kHD + h * kDH + lrow);
#pragma unroll
        for (int t = 0; t < 4; ++t) {
          v2f bf;
          bf.x = vrow[t * 16];
          bf.y = vrow[t * 16 + kHD];
          ot[t] = __builtin_amdgcn_wmma_f32_16x16x4_f32(
              false, af, false, bf, (short)0, ot[t], false, false);
        }
      }
    }

    s_f[h][lrow] = 1.0f / denom;
    float nf[8];
#pragma unroll
    for (int r = 0; r < 8; ++r) nf[r] = s_f[h][r + lhi * 8];
#pragma unroll
    for (int t = 0; t < 4; ++t)
#pragma unroll
      for (int r = 0; r < 8; ++r) res[t][r] += ot[t][r] * nf[r];
  }

#pragma unroll
  for (int t = 0; t < 4; ++t)
#pragma unroll
    for (int r = 0; r < 8; ++r)
      og[(size_t)(bI * kN + n0 + r + lhi * 8) * kHD + h * kDH + t * 16 + lrow] =
          res[t][r];
}

// ---------------------------------------------------------------------------
extern "C" void kernel_launch(void* const* d_in, const int* in_sizes, int n_in,
                              void* d_out, int out_size, void* d_ws, size_t ws_size,
                              hipStream_t stream) {
  const float* x    = (const float*)d_in[0];
  const float* ctx  = (const float*)d_in[1];
  const float* adp  = (const float*)d_in[2];
  const float* Wq   = (const float*)d_in[3];
  const float* Wk   = (const float*)d_in[4];
  const float* Wv   = (const float*)d_in[5];
  const float* Wo   = (const float*)d_in[6];
  const float* bo   = (const float*)d_in[7];

  float* out   = (float*)d_out;                          // [B,N,QD]
  float* probs = out + (size_t)kB * kN * kQD;            // [B,N,M]

  float* qb = (float*)d_ws;                              // [B,N,HD]   16 MB
  float* kb = qb + (size_t)kB * kN * kHD;                // [B,M,HD]    4 MB
  float* vb = kb + (size_t)kB * kM * kHD;                // [B,M,HD]    4 MB
  float* ab = vb + (size_t)kB * kM * kHD;                // [B,N,HD]   16 MB

  auto launch_gemm = [&](const float* A, const float* W, const float* bias,
                         float* C, int Mr, int K, int Nn) {
    const int blocks = (Mr / 128) * (Nn / 64);           // exact tiling
    gemm_wmma_f32<<<blocks, 256, 0, stream>>>(A, W, bias, C, Mr, K, Nn);
  };

  launch_gemm(x,   Wq, nullptr, qb, kB * kN, kQD, kHD);  // q = x @ Wq
  launch_gemm(ctx, Wk, nullptr, kb, kB * kM, kCD, kHD);  // k = ctx @ Wk
  launch_gemm(ctx, Wv, nullptr, vb, kB * kM, kCD, kHD);  // v = ctx @ Wv

  attn_fused<<<dim3(kN / 16, kB), 256, 0, stream>>>(qb, adp, kb, vb, ab, probs);

  launch_gemm(ab, Wo, bo, out, kB * kN, kHD, kQD);
}